// HomoGCN_11914239279715
// MI455X (gfx1250) — compile-verified
//
#include <hip/hip_runtime.h>
#include <hip/hip_bf16.h>

typedef float v2f __attribute__((ext_vector_type(2)));
typedef float v8f __attribute__((ext_vector_type(8)));

#define D 64
#define BN_EPS 1e-5f
#define SCAN_B 256

// ---------------------------------------------------------------------------
// Zero an int buffer
// ---------------------------------------------------------------------------
__global__ void zero_i32_kernel(int* __restrict__ p, int n) {
    int i = blockIdx.x * blockDim.x + threadIdx.x;
    if (i < n) p[i] = 0;
}

// ---------------------------------------------------------------------------
// In-degree histogram: degi[col[e]] += 1
// ---------------------------------------------------------------------------
__global__ void degree_kernel(const long long* __restrict__ col,
                              int* __restrict__ degi, int E) {
    int e = blockIdx.x * blockDim.x + threadIdx.x;
    if (e < E) atomicAdd(&degi[(int)col[e]], 1);
}

// ---------------------------------------------------------------------------
// dis[n] = deg>0 ? rsqrt(deg) : 0
// ---------------------------------------------------------------------------
__global__ void dis_kernel(const int* __restrict__ degi,
                           float* __restrict__ dis, int N) {
    int n = blockIdx.x * blockDim.x + threadIdx.x;
    if (n < N) {
        int d = degi[n];
        dis[n] = (d > 0) ? rsqrtf((float)d) : 0.0f;
    }
}

// ---------------------------------------------------------------------------
// Block-level exclusive scan (Hillis-Steele in LDS); emits per-block totals.
// ---------------------------------------------------------------------------
__global__ void scan_block_kernel(const int* __restrict__ in,
                                  int* __restrict__ out,
                                  int* __restrict__ bsums, int n) {
    __shared__ int s[SCAN_B];
    const int t = threadIdx.x;
    const int i = blockIdx.x * SCAN_B + t;
    int v = (i < n) ? in[i] : 0;
    s[t] = v;
    __syncthreads();
#pragma unroll
    for (int o = 1; o < SCAN_B; o <<= 1) {
        int u = (t >= o) ? s[t - o] : 0;
        __syncthreads();
        s[t] += u;
        __syncthreads();
    }
    int incl = s[t];
    if (i < n) out[i] = incl - v;            // exclusive
    if (t == SCAN_B - 1) bsums[blockIdx.x] = incl;
}

// ---------------------------------------------------------------------------
// Single-block exclusive scan over the block sums (looped with carry).
// ---------------------------------------------------------------------------
__global__ void scan_bsums_kernel(int* __restrict__ bs, int nb) {
    __shared__ int s[SCAN_B];
    const int t = threadIdx.x;
    int carry = 0;
    for (int base = 0; base < nb; base += SCAN_B) {
        int i = base + t;
        int v = (i < nb) ? bs[i] : 0;
        s[t] = v;
        __syncthreads();
#pragma unroll
        for (int o = 1; o < SCAN_B; o <<= 1) {
            int u = (t >= o) ? s[t - o] : 0;
            __syncthreads();
            s[t] += u;
            __syncthreads();
        }
        int incl = s[t];
        if (i < nb) bs[i] = carry + incl - v;  // exclusive
        int total = s[SCAN_B - 1];
        __syncthreads();
        carry += total;
    }
}

// ---------------------------------------------------------------------------
// Add block offsets; also seal off[N] = E.
// ---------------------------------------------------------------------------
__global__ void scan_add_kernel(int* __restrict__ off,
                                const int* __restrict__ bsums, int n, int E) {
    int i = blockIdx.x * SCAN_B + threadIdx.x;
    if (i < n) off[i] += bsums[blockIdx.x];
    if (i == 0) off[n] = E;
}

// ---------------------------------------------------------------------------
// cursor = off   (fresh fill cursors every call -> replay-deterministic)
// ---------------------------------------------------------------------------
__global__ void copy_i32_kernel(const int* __restrict__ src,
                                int* __restrict__ dst, int n) {
    int i = blockIdx.x * blockDim.x + threadIdx.x;
    if (i < n) dst[i] = src[i];
}

// ---------------------------------------------------------------------------
// CSR fill: slot = cursor[col]++; csr_src[slot]=row; csr_w[slot]=dis_r*dis_c
// ---------------------------------------------------------------------------
__global__ void fill_kernel(const long long* __restrict__ row,
                            const long long* __restrict__ col,
                            const float* __restrict__ dis,
                            int* __restrict__ cursor,
                            int* __restrict__ csr_src,
                            float* __restrict__ csr_w, int E) {
    int e = blockIdx.x * blockDim.x + threadIdx.x;
    if (e < E) {
        int r = (int)row[e];
        int c = (int)col[e];
        int pos = atomicAdd(&cursor[c], 1);
        csr_src[pos] = r;
        csr_w[pos]   = dis[r] * dis[c];
    }
}

// ---------------------------------------------------------------------------
// Dense GEMM  out[N,64] = H[N,64] @ Wm[64,64]  via V_WMMA_F32_16X16X4_F32.
// One wave per 16x16 output tile; B-panel in registers; 16 chained WMMAs.
// ---------------------------------------------------------------------------
__global__ void gemm_wmma_f32_kernel(const float* __restrict__ H,
                                     const float* __restrict__ Wm,
                                     float* __restrict__ out, int N) {
    const int lane  = threadIdx.x & 31;
    const int wave  = (blockIdx.x * blockDim.x + threadIdx.x) >> 5;
    const int ntile = wave & 3;
    const int mtile = wave >> 2;
    const int mbase = mtile * 16;
    if (mbase >= N) return;

    const int half = lane >> 4;
    const int l15  = lane & 15;
    const int ncol = ntile * 16 + l15;

    v2f bfrag[16];
#pragma unroll
    for (int s = 0; s < 16; ++s) {
        int k = s * 4 + half * 2;
        bfrag[s].x = Wm[(k + 0) * D + ncol];
        bfrag[s].y = Wm[(k + 1) * D + ncol];
    }

    int arow = mbase + l15;
    if (arow >= N) arow = N - 1;
    const float* hrow = H + (long long)arow * D + half * 2;

    v8f acc = {};
#pragma unroll
    for (int s = 0; s < 16; ++s) {
        v2f afrag;
        afrag.x = hrow[s * 4 + 0];
        afrag.y = hrow[s * 4 + 1];
        acc = __builtin_amdgcn_wmma_f32_16x16x4_f32(
            false, afrag, false, bfrag[s], (short)0, acc, false, false);
    }

#pragma unroll
    for (int j = 0; j < 8; ++j) {
        int r = mbase + j + half * 8;
        if (r < N) out[(long long)r * D + ncol] = acc[j];
    }
}

// ---------------------------------------------------------------------------
// CSR gather-aggregate + bias + BatchNorm(eval) + ReLU, fused.
// 16 threads per node; each owns a float4 feature quad (global_load_b128).
// Zero atomics; edge-list scalars are shared by the 16 lanes of a node (two nodes per wave).
// ---------------------------------------------------------------------------
__global__ void gather_bn_relu_kernel(const float* __restrict__ hw,
                                      const int* __restrict__ off,
                                      const int* __restrict__ csr_src,
                                      const float* __restrict__ csr_w,
                                      const float* __restrict__ bias,
                                      const float* __restrict__ gamma,
                                      const float* __restrict__ beta,
                                      const float* __restrict__ rmean,
                                      const float* __restrict__ rvar,
                                      float* __restrict__ dst, int N) {
    long long t = (long long)blockIdx.x * blockDim.x + threadIdx.x;
    int node = (int)(t >> 4);
    int q    = (int)(t & 15) * 4;          // feature quad base
    if (node >= N) return;

    const int s0 = off[node];
    const int s1 = off[node + 1];

    float a0 = 0.f, a1 = 0.f, a2 = 0.f, a3 = 0.f;
    for (int j = s0; j < s1; ++j) {
        float w = csr_w[j];
        const float4 h4 = *(const float4*)(hw + (long long)csr_src[j] * D + q);
        a0 += w * h4.x;
        a1 += w * h4.y;
        a2 += w * h4.z;
        a3 += w * h4.w;
    }

    float4 o;
    {
        float v = a0 + bias[q + 0];
        v = (v - rmean[q + 0]) * rsqrtf(rvar[q + 0] + BN_EPS) * gamma[q + 0] + beta[q + 0];
        o.x = fmaxf(v, 0.f);
        v = a1 + bias[q + 1];
        v = (v - rmean[q + 1]) * rsqrtf(rvar[q + 1] + BN_EPS) * gamma[q + 1] + beta[q + 1];
        o.y = fmaxf(v, 0.f);
        v = a2 + bias[q + 2];
        v = (v - rmean[q + 2]) * rsqrtf(rvar[q + 2] + BN_EPS) * gamma[q + 2] + beta[q + 2];
        o.z = fmaxf(v, 0.f);
        v = a3 + bias[q + 3];
        v = (v - rmean[q + 3]) * rsqrtf(rvar[q + 3] + BN_EPS) * gamma[q + 3] + beta[q + 3];
        o.w = fmaxf(v, 0.f);
    }
    *(float4*)(dst + (long long)node * D + q) = o;
}

// ---------------------------------------------------------------------------
static inline long long align256(long long x) { return (x + 255) & ~255LL; }

extern "C" void kernel_launch(void* const* d_in, const int* in_sizes, int n_in,
                              void* d_out, int out_size, void* d_ws, size_t ws_size,
                              hipStream_t stream) {
    const float*     x   = (const float*)d_in[0];
    const long long* ei  = (const long long*)d_in[1];   // int64 [2,E]
    const float*     Wl  = (const float*)d_in[2];       // [L,64,64]
    const float*     bl  = (const float*)d_in[3];       // [L,64]
    const float*     gl  = (const float*)d_in[4];
    const float*     bel = (const float*)d_in[5];
    const float*     rml = (const float*)d_in[6];
    const float*     rvl = (const float*)d_in[7];
    float*           out = (float*)d_out;

    const int N = in_sizes[0] / D;
    const int E = in_sizes[1] / 2;
    const int L = in_sizes[2] / (D * D);

    const long long* row = ei;       // src
    const long long* col = ei + E;   // dst

    // Workspace carve-up
    char* ws = (char*)d_ws;
    long long o = 0;
    int*   degi    = (int*)(ws + o);   o += align256((long long)N * 4);
    int*   off     = (int*)(ws + o);   o += align256((long long)(N + 1) * 4);
    int*   cursor  = (int*)(ws + o);   o += align256((long long)N * 4);
    int*   bsums   = (int*)(ws + o);   o += align256((long long)(N / SCAN_B + 2) * 4);
    int*   csr_src = (int*)(ws + o);   o += align256((long long)E * 4);
    float* csr_w   = (float*)(ws + o); o += align256((long long)E * 4);
    float* dis     = (float*)(ws + o); o += align256((long long)N * 4);
    float* bufA    = (float*)(ws + o); o += align256((long long)N * D * 4);
    float* bufB    = (float*)(ws + o); o += align256((long long)N * D * 4);
    float* bufC    = (float*)(ws + o); o += align256((long long)N * D * 4);
    (void)ws_size;

    const int TB = 256;
    const int nB  = (N + TB - 1) / TB;   // node-grid
    const int eB  = (E + TB - 1) / TB;   // edge-grid
    const int nbScan = (N + SCAN_B - 1) / SCAN_B;

    // --- gcn_norm + CSR build (once; shared by all layers) ---
    zero_i32_kernel<<<nB, TB, 0, stream>>>(degi, N);
    degree_kernel<<<eB, TB, 0, stream>>>(col, degi, E);
    dis_kernel<<<nB, TB, 0, stream>>>(degi, dis, N);
    scan_block_kernel<<<nbScan, SCAN_B, 0, stream>>>(degi, off, bsums, N);
    scan_bsums_kernel<<<1, SCAN_B, 0, stream>>>(bsums, nbScan);
    scan_add_kernel<<<nbScan, SCAN_B, 0, stream>>>(off, bsums, N, E);
    copy_i32_kernel<<<nB, TB, 0, stream>>>(off, cursor, N);
    fill_kernel<<<eB, TB, 0, stream>>>(row, col, dis, cursor, csr_src, csr_w, E);

    // GEMM geometry: one wave per 16x16 tile, 4 column tiles
    const long long gemm_threads = (long long)((N + 15) / 16) * 4 * 32;
    const int gemm_blocks = (int)((gemm_threads + TB - 1) / TB);

    // Gather geometry: 16 threads per node
    const long long gat_threads = (long long)N * 16;
    const int gat_blocks = (int)((gat_threads + TB - 1) / TB);

    const float* cur = x;
    float* ping = bufB;
    float* pong = bufC;
    for (int l = 0; l < L; ++l) {
        gemm_wmma_f32_kernel<<<gemm_blocks, TB, 0, stream>>>(
            cur, Wl + (long long)l * D * D, bufA, N);

        float* dst = (l == L - 1) ? out : ping;
        gather_bn_relu_kernel<<<gat_blocks, TB, 0, stream>>>(
            bufA, off, csr_src, csr_w,
            bl + l * D, gl + l * D, bel + l * D, rml + l * D, rvl + l * D,
            dst, N);

        cur = dst;
        float* tswap = ping; ping = pong; pong = tswap;
    }
}